// GVIAttentionBlock_74096775791194
// MI455X (gfx1250) — compile-verified
//
#include <hip/hip_runtime.h>
#include <cmath>

// ---------------------------------------------------------------------------
// GVIAttentionBlock for MI455X (gfx1250, wave32).
//   B=32, N=65536, H=6, CIN=64, DF=1024
// Bandwidth-bound: x is 48MB, out 48MB. Two streaming passes over x (second
// pass should hit the 192MB L2) + tiny WMMA GEMM stage for f/g matrices.
// ---------------------------------------------------------------------------

#define B_    32
#define N_    65536
#define H_    6
#define CIN_  64
#define DF_   1024
#define EPS_  1e-5f
#define MAXLS 4.6051701859880913680f   // log(1/0.01)

#define SCHUNKS 32     // chunks per batch for the streaming passes
#define TPB     256    // threads per block (8 waves)
#define TOKS    8      // tokens per thread (48 contiguous floats = 12 float4)

// workspace layout (floats)
#define G_OFF   0                          // g[b][h][o] scale-folded, B*36
#define P_OFF   (B_*36)                    // partials: B*SCHUNKS*12 (max6,sum6)
#define S_OFF   (P_OFF + B_*SCHUNKS*12)    // stats: B*12 (max6, invdenom6)

typedef __attribute__((ext_vector_type(2))) float v2f;
typedef __attribute__((ext_vector_type(8))) float v8f;

// ---------------------------------------------------------------------------
// Stage 1: per batch, compute g[h][o] = (bc[h] + Wc[h,:]·F1[:,o]) * scale[o]
//   H  = relu(feature[b] @ W1^T + b1)        [64 x 6]   <- WMMA f32 16x16x4
//   F1 = H @ W2^T + b2                       [64 x 6]
//   g  = (Wc @ F1 + bc) * exp(min(ls, MAX))  [ 6 x 6]
// One block per batch, 128 threads = 4 waves; wave w owns rows 16w..16w+15.
// ---------------------------------------------------------------------------
__global__ __launch_bounds__(128)
void stage1_kernel(const float* __restrict__ feature, const float* __restrict__ W1,
                   const float* __restrict__ b1, const float* __restrict__ W2,
                   const float* __restrict__ b2, const float* __restrict__ Wc,
                   const float* __restrict__ bc, const float* __restrict__ logit_scale,
                   float* __restrict__ g)
{
    __shared__ float Hs[CIN_ * H_];
    __shared__ float F1[CIN_ * H_];

    const int b    = blockIdx.x;
    const int tid  = threadIdx.x;
    const int wave = tid >> 5;
    const int lane = tid & 31;
    const int hb   = lane >> 4;     // lane-half: selects K pair {0,1} or {2,3}
    const int ln   = lane & 15;     // M (A rows) / N (B cols) within tile

    const int  row    = wave * 16 + ln;                 // feature row c (0..63)
    const bool nvalid = (ln < H_);
    const float msk   = nvalid ? 1.0f : 0.0f;

    const float* frow = feature + ((size_t)b * CIN_ + row) * DF_;
    const float* wrow = W1 + (size_t)(nvalid ? ln : 0) * DF_;  // clamped: no OOB

    // A 16x4 f32 layout: lane<16 -> K={0,1} in v0/v1 ; lane>=16 -> K={2,3}.
    // B 4x16 f32 layout mirrors with N across lanes.  K = 2*hb + vgpr.
    v8f acc = {};
    #pragma unroll 4
    for (int k = 0; k < DF_; k += 4) {
        const int kk = k + 2 * hb;
        v2f a;  a.x = frow[kk];          a.y = frow[kk + 1];
        v2f bb; bb.x = wrow[kk] * msk;   bb.y = wrow[kk + 1] * msk;
        acc = __builtin_amdgcn_wmma_f32_16x16x4_f32(
                  false, a, false, bb, (short)0, acc, false, false);
    }

    // C layout: VGPR r -> M = r + 8*hb, N = ln.  Apply bias + ReLU, stash tile.
    if (nvalid) {
        const float bias = b1[ln];
        #pragma unroll
        for (int r = 0; r < 8; ++r) {
            const int c = wave * 16 + hb * 8 + r;
            Hs[c * H_ + ln] = fmaxf(acc[r] + bias, 0.0f);
        }
    }
    __syncthreads();

    // F1[c][o] = b2[o] + sum_h H[c][h] * W2[o][h]
    for (int idx = tid; idx < CIN_ * H_; idx += 128) {
        const int c = idx / H_, o = idx % H_;
        float v = b2[o];
        #pragma unroll
        for (int h2 = 0; h2 < H_; ++h2) v += Hs[c * H_ + h2] * W2[o * H_ + h2];
        F1[idx] = v;
    }
    __syncthreads();

    // g[h][o] = (bc[h] + sum_c Wc[h][c] * F1[c][o]) * exp(min(ls[o], MAXLS))
    if (tid < H_ * H_) {
        const int h = tid / H_, o = tid % H_;
        float v = bc[h];
        for (int c = 0; c < CIN_; ++c) v += Wc[h * CIN_ + c] * F1[c * H_ + o];
        const float scale = __expf(fminf(logit_scale[o], MAXLS));
        g[b * 36 + h * H_ + o] = v * scale;
    }
}

// ---------------------------------------------------------------------------
// Pass 1: streaming softmax statistics.  t[n][o] = sum_h x[n][h]*g[h][o].
// Each block: 2048 contiguous tokens of one batch; per-thread 8 tokens
// (12 aligned float4 loads).  LDS tree-reduce (max, sumexp) per head.
// ---------------------------------------------------------------------------
__global__ __launch_bounds__(TPB)
void pass1_kernel(const float* __restrict__ x, const float* __restrict__ g,
                  float* __restrict__ part)
{
    __shared__ float gs[36];
    __shared__ float red[TPB * 12];

    const int b     = blockIdx.y;
    const int chunk = blockIdx.x;
    const int tid   = threadIdx.x;
    if (tid < 36) gs[tid] = g[b * 36 + tid];
    __syncthreads();

    const size_t tok0 = (size_t)b * N_ + (size_t)chunk * (TPB * TOKS)
                      + (size_t)tid * TOKS;
    const float4* px = (const float4*)(x + tok0 * H_);

    float v[TOKS * H_];
    #pragma unroll
    for (int i = 0; i < TOKS * H_ / 4; ++i) {
        float4 q = px[i];
        v[4*i] = q.x; v[4*i+1] = q.y; v[4*i+2] = q.z; v[4*i+3] = q.w;
    }

    float mx[H_];
    #pragma unroll
    for (int o = 0; o < H_; ++o) mx[o] = -1e30f;

    #pragma unroll
    for (int j = 0; j < TOKS; ++j) {
        float t[H_];
        #pragma unroll
        for (int o = 0; o < H_; ++o) {
            float a = 0.f;
            #pragma unroll
            for (int h = 0; h < H_; ++h) a += v[j*H_ + h] * gs[h*H_ + o];
            t[o] = a;
        }
        #pragma unroll
        for (int o = 0; o < H_; ++o) {       // overwrite x with t (regs reused)
            v[j*H_ + o] = t[o];
            mx[o] = fmaxf(mx[o], t[o]);
        }
    }

    float sm[H_];
    #pragma unroll
    for (int o = 0; o < H_; ++o) sm[o] = 0.f;
    #pragma unroll
    for (int j = 0; j < TOKS; ++j)
        #pragma unroll
        for (int o = 0; o < H_; ++o) sm[o] += __expf(v[j*H_ + o] - mx[o]);

    #pragma unroll
    for (int o = 0; o < H_; ++o) { red[tid*12 + o] = mx[o]; red[tid*12 + 6 + o] = sm[o]; }
    __syncthreads();

    for (int off = TPB / 2; off > 0; off >>= 1) {
        if (tid < off) {
            #pragma unroll
            for (int o = 0; o < H_; ++o) {
                float m1 = red[tid*12 + o],        s1 = red[tid*12 + 6 + o];
                float m2 = red[(tid+off)*12 + o],  s2 = red[(tid+off)*12 + 6 + o];
                float M  = fmaxf(m1, m2);
                red[tid*12 + o]     = M;
                red[tid*12 + 6 + o] = s1 * __expf(m1 - M) + s2 * __expf(m2 - M);
            }
        }
        __syncthreads();
    }
    if (tid < 12)
        part[((size_t)b * gridDim.x + chunk) * 12 + tid] = red[tid];
}

// ---------------------------------------------------------------------------
// Reduce per-block partials -> per (b,o): global max M and 1/denominator.
// One thread per (b,o): 32*6 = 192 threads, single block.
// ---------------------------------------------------------------------------
__global__ __launch_bounds__(192)
void reduce_kernel(const float* __restrict__ part, float* __restrict__ stats)
{
    const int tid = threadIdx.x;          // 0..191
    const int b = tid / H_, o = tid % H_;
    float M = -1e30f;
    for (int j = 0; j < SCHUNKS; ++j)
        M = fmaxf(M, part[((size_t)b * SCHUNKS + j) * 12 + o]);
    float S = 0.f;
    for (int j = 0; j < SCHUNKS; ++j) {
        const float m = part[((size_t)b * SCHUNKS + j) * 12 + o];
        const float s = part[((size_t)b * SCHUNKS + j) * 12 + 6 + o];
        S += s * __expf(m - M);
    }
    stats[b * 12 + o]      = M;
    stats[b * 12 + 6 + o]  = 1.0f / S;
}

// ---------------------------------------------------------------------------
// Pass 2: re-stream x (L2-hot), softmax-normalize, gate by x, LayerNorm(6).
// ---------------------------------------------------------------------------
__global__ __launch_bounds__(TPB)
void pass2_kernel(const float* __restrict__ x, const float* __restrict__ g,
                  const float* __restrict__ stats, const float* __restrict__ gamma,
                  const float* __restrict__ beta, float* __restrict__ out)
{
    __shared__ float gs[36];
    __shared__ float st[12];
    __shared__ float gb[12];

    const int b     = blockIdx.y;
    const int chunk = blockIdx.x;
    const int tid   = threadIdx.x;
    if (tid < 36)                 gs[tid]        = g[b * 36 + tid];
    if (tid >= 64 && tid < 76)    st[tid - 64]   = stats[b * 12 + (tid - 64)];
    if (tid >= 96 && tid < 102)   gb[tid - 96]   = gamma[tid - 96];
    if (tid >= 102 && tid < 108)  gb[tid - 96]   = beta[tid - 102];
    __syncthreads();

    const size_t tok0 = (size_t)b * N_ + (size_t)chunk * (TPB * TOKS)
                      + (size_t)tid * TOKS;
    const float4* px   = (const float4*)(x + tok0 * H_);
    float4*       pout = (float4*)(out + tok0 * H_);

    float v[TOKS * H_];
    #pragma unroll
    for (int i = 0; i < TOKS * H_ / 4; ++i) {
        float4 q = px[i];
        v[4*i] = q.x; v[4*i+1] = q.y; v[4*i+2] = q.z; v[4*i+3] = q.w;
    }

    #pragma unroll
    for (int j = 0; j < TOKS; ++j) {
        float t[H_], y[H_];
        #pragma unroll
        for (int o = 0; o < H_; ++o) {
            float a = 0.f;
            #pragma unroll
            for (int h = 0; h < H_; ++h) a += v[j*H_ + h] * gs[h*H_ + o];
            t[o] = a;
        }
        float mu = 0.f;
        #pragma unroll
        for (int o = 0; o < H_; ++o) {
            y[o] = __expf(t[o] - st[o]) * st[6 + o] * v[j*H_ + o];
            mu += y[o];
        }
        mu *= (1.0f / H_);
        float var = 0.f;
        #pragma unroll
        for (int o = 0; o < H_; ++o) { const float d = y[o] - mu; var += d * d; }
        var *= (1.0f / H_);
        const float inv = rsqrtf(var + EPS_);
        #pragma unroll
        for (int o = 0; o < H_; ++o)
            v[j*H_ + o] = (y[o] - mu) * inv * gb[o] + gb[6 + o];
    }

    #pragma unroll
    for (int i = 0; i < TOKS * H_ / 4; ++i) {
        float4 q; q.x = v[4*i]; q.y = v[4*i+1]; q.z = v[4*i+2]; q.w = v[4*i+3];
        pout[i] = q;
    }
}

// ---------------------------------------------------------------------------
extern "C" void kernel_launch(void* const* d_in, const int* in_sizes, int n_in,
                              void* d_out, int out_size, void* d_ws, size_t ws_size,
                              hipStream_t stream)
{
    const float* x       = (const float*)d_in[0];
    const float* feature = (const float*)d_in[1];
    const float* W1      = (const float*)d_in[2];
    const float* b1      = (const float*)d_in[3];
    const float* W2      = (const float*)d_in[4];
    const float* b2      = (const float*)d_in[5];
    const float* Wc      = (const float*)d_in[6];
    const float* bc      = (const float*)d_in[7];
    const float* ls      = (const float*)d_in[8];
    const float* gamma   = (const float*)d_in[9];
    const float* beta    = (const float*)d_in[10];
    float* out = (float*)d_out;

    float* ws    = (float*)d_ws;
    float* g     = ws + G_OFF;
    float* part  = ws + P_OFF;
    float* stats = ws + S_OFF;

    stage1_kernel<<<B_, 128, 0, stream>>>(feature, W1, b1, W2, b2, Wc, bc, ls, g);

    dim3 grid(SCHUNKS, B_);
    pass1_kernel<<<grid, TPB, 0, stream>>>(x, g, part);
    reduce_kernel<<<1, 192, 0, stream>>>(part, stats);
    pass2_kernel<<<grid, TPB, 0, stream>>>(x, g, stats, gamma, beta, out);
}